// AttLayer_8770323218764
// MI455X (gfx1250) — compile-verified
//
#include <hip/hip_runtime.h>
#include <math.h>

// ---------------------------------------------------------------------------
// Few-shot attention pipeline for MI455X (gfx1250), fp32 via V_WMMA_F32_16X16X4_F32.
// All GEMM outputs stored channel-innermost (b, p, c) so every WMMA operand
// is a contiguous float2 (b64) load per lane.
// Register blocking: each wave computes 4 adjacent 16x16 tiles sharing the
// A operand -> per K-step: 1 A load + 4 B loads + 4 independent WMMAs.
// Attention max-reduction is kept per-lane in registers and collapsed with a
// single cross-lane shuffle reduction at the end of the key sweep.
// ---------------------------------------------------------------------------

#define N_WAY 5
#define N_SHOT 5
#define CC 640
#define HWP 1024
#define BB 25
#define QTOT 5120          // N_SHOT*HW
#define EPSF 1e-12f
#define SCALE_F 0.03952847075210474f   // 1/sqrt(640)

typedef float v2f __attribute__((ext_vector_type(2)));
typedef float v8f __attribute__((ext_vector_type(8)));

// ---------------- GEMM: Out[b,p,c_out] = W*In + bias (+ residual) ----------
// One wave: 16 out-channels x 64 pixels (4 tiles of 16x16).
// IN_CP: In is (b, c, p) [raw x5];  !IN_CP: In is (b, p, c).
template<bool IN_CP, bool HAS_RES>
__global__ __launch_bounds__(256) void k_gemm(
    const float* __restrict__ W, const float* __restrict__ bias,
    const float* __restrict__ In, const float* __restrict__ Res,
    float* __restrict__ Out)
{
    const int lane = threadIdx.x & 31;
    const int wid  = threadIdx.x >> 5;
    const int TO = CC / 16, TPG = HWP / 64;         // 40, 16
    int t  = blockIdx.x * 8 + wid;                  // < 25*40*16 = 16000
    int b  = t / (TO * TPG);
    int r2 = t % (TO * TPG);
    int o0 = (r2 / TPG) * 16;
    int pg = (r2 % TPG) * 64;
    const bool hi = lane >= 16;
    const int  lm = lane & 15;
    const int  m0 = o0 + (hi ? 8 : 0);

    v8f acc[4];
#pragma unroll
    for (int j = 0; j < 4; ++j)
#pragma unroll
        for (int r = 0; r < 8; ++r) acc[j][r] = bias[m0 + r];

    const float* Wrow = W + (size_t)(o0 + lm) * CC;
    const float* Bp[4];
#pragma unroll
    for (int j = 0; j < 4; ++j) {
        const int n = pg + j * 16 + lm;
        Bp[j] = IN_CP ? (In + (size_t)b * CC * HWP + n)       // (b,c,p): +kk*HWP
                      : (In + ((size_t)b * HWP + n) * CC);    // (b,p,c): +kk
    }

#pragma unroll 2
    for (int k = 0; k < CC; k += 4) {
        const int kk = k + (hi ? 2 : 0);
        v2f a = *(const v2f*)(Wrow + kk);
#pragma unroll
        for (int j = 0; j < 4; ++j) {
            v2f bv;
            if (IN_CP) {
                bv[0] = Bp[j][(size_t)kk * HWP];
                bv[1] = Bp[j][(size_t)(kk + 1) * HWP];
            } else {
                bv = *(const v2f*)(Bp[j] + kk);
            }
            acc[j] = __builtin_amdgcn_wmma_f32_16x16x4_f32(
                false, a, false, bv, (short)0, acc[j], false, false);
        }
    }

#pragma unroll
    for (int j = 0; j < 4; ++j) {
        const int n = pg + j * 16 + lm;
        float* outp = Out + ((size_t)b * HWP + n) * CC + m0;
        if (HAS_RES) {
            const float* rp = Res + (size_t)b * CC * HWP + n;
#pragma unroll
            for (int r = 0; r < 8; ++r)
                outp[r] = acc[j][r] + rp[(size_t)(m0 + r) * HWP];
        } else {
#pragma unroll
            for (int r = 0; r < 8; ++r) outp[r] = acc[j][r];
        }
    }
}

// ------------- attention scores, fused max over key pixels ----------------
// One wave: 16 queries x one key-shot; key tiles processed 4 at a time
// sharing the A (query) operand. Per-lane register max across all tiles;
// one cross-lane shuffle reduction at the end.
// Writes red[(w*5120+q)*5+sk] = scale * max_pk score.
__global__ __launch_bounds__(256) void k_attn(
    const float* __restrict__ XQ, const float* __restrict__ XK,
    float* __restrict__ red)
{
    const int lane = threadIdx.x & 31;
    const int wid  = threadIdx.x >> 5;
    int t  = blockIdx.x * 8 + wid;                  // < 5*5*320
    int qt = t % 320;
    int ws = t / 320;
    int sk = ws % N_SHOT;
    int w  = ws / N_SHOT;
    const bool hi = lane >= 16;
    const int  lm = lane & 15;

    int q  = qt * 16 + lm;                          // 0..5119
    int sq = q >> 10, pq = q & 1023;
    const float* Aq = XQ + ((size_t)(w * N_SHOT + sq) * HWP + pq) * CC;
    const float* KB = XK + (size_t)(w * N_SHOT + sk) * HWP * CC;

    float rowmax[8];
#pragma unroll
    for (int r = 0; r < 8; ++r) rowmax[r] = -3.4e38f;

    for (int kt = 0; kt < 64; kt += 4) {
        const float* Bk[4];
#pragma unroll
        for (int j = 0; j < 4; ++j)
            Bk[j] = KB + (size_t)((kt + j) * 16 + lm) * CC;

        v8f acc[4];
#pragma unroll
        for (int j = 0; j < 4; ++j)
#pragma unroll
            for (int r = 0; r < 8; ++r) acc[j][r] = 0.0f;

#pragma unroll 2
        for (int k = 0; k < CC; k += 4) {
            const int kk = k + (hi ? 2 : 0);
            v2f a = *(const v2f*)(Aq + kk);
#pragma unroll
            for (int j = 0; j < 4; ++j) {
                v2f bv = *(const v2f*)(Bk[j] + kk);
                acc[j] = __builtin_amdgcn_wmma_f32_16x16x4_f32(
                    false, a, false, bv, (short)0, acc[j], false, false);
            }
        }
        // fold the 4 tiles into the per-lane running max (pure VALU)
#pragma unroll
        for (int r = 0; r < 8; ++r) {
            float m01 = fmaxf(acc[0][r], acc[1][r]);
            float m23 = fmaxf(acc[2][r], acc[3][r]);
            rowmax[r] = fmaxf(rowmax[r], fmaxf(m01, m23));
        }
    }

    // one cross-lane max reduction over the 16 key columns per half-wave
#pragma unroll
    for (int r = 0; r < 8; ++r) {
        float v = rowmax[r];
        v = fmaxf(v, __shfl_xor(v, 1, 16));
        v = fmaxf(v, __shfl_xor(v, 2, 16));
        v = fmaxf(v, __shfl_xor(v, 4, 16));
        v = fmaxf(v, __shfl_xor(v, 8, 16));
        rowmax[r] = v;
    }
    if (lm == 0) {
#pragma unroll
        for (int r = 0; r < 8; ++r) {
            int m = (hi ? 8 : 0) + r;
            int qr = qt * 16 + m;
            red[((size_t)w * QTOT + qr) * N_SHOT + sk] = rowmax[r] * SCALE_F;
        }
    }
}

// --------- mean over key shots + argmax-equality mask per (w,s) -----------
__global__ __launch_bounds__(256) void k_mask(
    const float* __restrict__ red, float* __restrict__ mask)
{
    const int ws = blockIdx.x;                      // w*5+s
    const int w = ws / N_SHOT, s = ws % N_SHOT;
    const int tid = threadIdx.x;
    __shared__ float sm[256];
    float v[4], lmax = -3.4e38f;
#pragma unroll
    for (int i = 0; i < 4; ++i) {
        int p = tid + i * 256;
        const float* rp = red + ((size_t)w * QTOT + s * HWP + p) * N_SHOT;
        v[i] = (rp[0] + rp[1] + rp[2] + rp[3] + rp[4]) * 0.2f;
        lmax = fmaxf(lmax, v[i]);
    }
    sm[tid] = lmax; __syncthreads();
    for (int st = 128; st > 0; st >>= 1) {
        if (tid < st) sm[tid] = fmaxf(sm[tid], sm[tid + st]);
        __syncthreads();
    }
    const float vmax = sm[0];
#pragma unroll
    for (int i = 0; i < 4; ++i)
        mask[(size_t)ws * HWP + tid + i * 256] = (v[i] == vmax) ? 1.0f : 0.0f;
}

// ---------------- per-pixel channel L2 norm of X --------------------------
__global__ __launch_bounds__(256) void k_norm(
    const float* __restrict__ X, float* __restrict__ nrm)
{
    const int lane = threadIdx.x & 31;
    const int wid  = threadIdx.x >> 5;
    size_t pix = (size_t)blockIdx.x * 8 + wid;      // < 25600
    const float* xp = X + pix * CC;
    float s = 0.0f;
    for (int c = lane; c < CC; c += 32) { float t = xp[c]; s += t * t; }
    s += __shfl_xor(s, 16); s += __shfl_xor(s, 8);
    s += __shfl_xor(s, 4);  s += __shfl_xor(s, 2);  s += __shfl_xor(s, 1);
    if (lane == 0) nrm[pix] = fmaxf(sqrtf(s), EPSF);
}

// ---------------- seeds = masked spatial sum of X/nrm ---------------------
__global__ void k_seeds(const float* __restrict__ X, const float* __restrict__ nrm,
                        const float* __restrict__ mask, float* __restrict__ seeds)
{
    const int b = blockIdx.x;                       // w*5+s
    const int c = threadIdx.x;                      // 640 threads
    float acc = 0.0f;
    for (int p = 0; p < HWP; ++p) {
        float m = mask[(size_t)b * HWP + p];
        if (m != 0.0f)
            acc += X[((size_t)b * HWP + p) * CC + c] / nrm[(size_t)b * HWP + p];
    }
    seeds[(size_t)b * CC + c] = acc;
}

// ------- cm[wb,o,p] = sum_{k,c} X[wb,k,p,c]/nrm * seeds[o,k,c] ------------
__global__ __launch_bounds__(256) void k_cm(
    const float* __restrict__ X, const float* __restrict__ nrm,
    const float* __restrict__ seeds, float* __restrict__ cm)
{
    const int lane = threadIdx.x & 31;
    const int wid  = threadIdx.x >> 5;
    const int pair = blockIdx.x >> 7;               // 0..24 : wb*5+o
    const int p = ((blockIdx.x & 127) << 3) + wid;  // 0..1023
    const int wb = pair / N_WAY, o = pair % N_WAY;
    float acc = 0.0f;
    for (int k = 0; k < N_SHOT; ++k) {
        const int bi = wb * N_SHOT + k;
        const float inv = 1.0f / nrm[(size_t)bi * HWP + p];
        const float* xp = X + ((size_t)bi * HWP + p) * CC;
        const float* sp = seeds + (size_t)(o * N_SHOT + k) * CC;
        float part = 0.0f;
        for (int c = lane; c < CC; c += 32) part += xp[c] * sp[c];
        acc += part * inv;
    }
    acc += __shfl_xor(acc, 16); acc += __shfl_xor(acc, 8);
    acc += __shfl_xor(acc, 4);  acc += __shfl_xor(acc, 2); acc += __shfl_xor(acc, 1);
    if (lane == 0) cm[(size_t)pair * HWP + p] = acc;
}

// ---------------- per-row min/max normalize of cm -------------------------
__global__ __launch_bounds__(256) void k_cmnorm(float* __restrict__ cm)
{
    const int row = blockIdx.x, tid = threadIdx.x;
    __shared__ float smn[256], smx[256];
    float v[4], mn = 3.4e38f, mx = -3.4e38f;
#pragma unroll
    for (int i = 0; i < 4; ++i) {
        v[i] = cm[(size_t)row * HWP + tid + i * 256];
        mn = fminf(mn, v[i]); mx = fmaxf(mx, v[i]);
    }
    smn[tid] = mn; smx[tid] = mx; __syncthreads();
    for (int st = 128; st > 0; st >>= 1) {
        if (tid < st) {
            smn[tid] = fminf(smn[tid], smn[tid + st]);
            smx[tid] = fmaxf(smx[tid], smx[tid + st]);
        }
        __syncthreads();
    }
    const float rmn = smn[0];
    const float inv = 1.0f / (smx[0] - rmn + EPSF);
#pragma unroll
    for (int i = 0; i < 4; ++i)
        cm[(size_t)row * HWP + tid + i * 256] = (v[i] - rmn) * inv;
}

// ---------------- proto[w,c] = mean_{s,p} X * cm --------------------------
__global__ void k_proto(const float* __restrict__ X, const float* __restrict__ cm,
                        float* __restrict__ out)
{
    const int w = blockIdx.x, c = threadIdx.x;      // 640 threads
    float acc = 0.0f;
    for (int s = 0; s < N_SHOT; ++s) {
        const int bi = w * N_SHOT + s;
        const float* cr = cm + (size_t)bi * HWP;
        const float* xb = X + (size_t)bi * HWP * CC + c;
        for (int p = 0; p < HWP; ++p) acc += xb[(size_t)p * CC] * cr[p];
    }
    out[(size_t)w * CC + c] = acc * (1.0f / 5120.0f);
}

// ---------------------------------------------------------------------------
extern "C" void kernel_launch(void* const* d_in, const int* in_sizes, int n_in,
                              void* d_out, int out_size, void* d_ws, size_t ws_size,
                              hipStream_t stream)
{
    (void)in_sizes; (void)n_in; (void)out_size; (void)ws_size;
    const float* x5 = (const float*)d_in[0];
    const float* Wc = (const float*)d_in[1];
    const float* bc = (const float*)d_in[2];
    const float* Wq = (const float*)d_in[3];
    const float* bq = (const float*)d_in[4];
    const float* Wk = (const float*)d_in[5];
    const float* bk = (const float*)d_in[6];
    float* out = (float*)d_out;

    const size_t BIG = (size_t)BB * HWP * CC;       // 16,384,000 floats
    float* ws    = (float*)d_ws;
    float* T1    = ws;                              // stage1 out, later XQ
    float* X     = ws + BIG;
    float* XK    = ws + 2 * BIG;
    float* nrm   = ws + 3 * BIG;                    // 25600
    float* red   = nrm + 25600;                     // 5*5120*5 = 128000
    float* mask  = red + 128000;                    // 25600
    float* seeds = mask + 25600;                    // 16000
    float* cm    = seeds + 16000;                   // 25600

    const int GEMM_BLOCKS = (BB * (CC / 16) * (HWP / 64)) / 8; // 2000

    k_gemm<true,  true ><<<GEMM_BLOCKS, 256, 0, stream>>>(Wc, bc, x5, x5, T1);
    k_gemm<false, false><<<GEMM_BLOCKS, 256, 0, stream>>>(Wq, bq, T1, nullptr, X);
    k_gemm<false, false><<<GEMM_BLOCKS, 256, 0, stream>>>(Wq, bq, X, nullptr, T1);
    k_gemm<false, false><<<GEMM_BLOCKS, 256, 0, stream>>>(Wk, bk, X, nullptr, XK);
    k_norm<<<3200, 256, 0, stream>>>(X, nrm);
    k_attn<<<1000, 256, 0, stream>>>(T1, XK, red);
    k_mask<<<25, 256, 0, stream>>>(red, mask);
    k_seeds<<<25, CC, 0, stream>>>(X, nrm, mask, seeds);
    k_cm<<<3200, 256, 0, stream>>>(X, nrm, seeds, cm);
    k_cmnorm<<<25, 256, 0, stream>>>(cm);
    k_proto<<<N_WAY, CC, 0, stream>>>(X, cm, out);
}